// GraphFusion_71399536328730
// MI455X (gfx1250) — compile-verified
//
#include <hip/hip_runtime.h>
#include <hip/hip_bf16.h>

typedef __attribute__((ext_vector_type(2))) float v2f;
typedef __attribute__((ext_vector_type(8))) float v8f;

#define HDIM 128

// ---------------- degree / normalization ----------------
__global__ void deg_init_kernel(float* deg, int n) {
    int gid = blockIdx.x * blockDim.x + threadIdx.x;
    if (gid < n) deg[gid] = 1.0f;              // self-loop contributes 1
}

__global__ void deg_accum_kernel(const int* __restrict__ col, float* deg, int e) {
    int gid = blockIdx.x * blockDim.x + threadIdx.x;
    if (gid < e) atomicAdd(&deg[col[gid]], 1.0f);
}

__global__ void deg_rsqrt_kernel(float* deg, int n) {
    int gid = blockIdx.x * blockDim.x + threadIdx.x;
    if (gid < n) deg[gid] = rsqrtf(deg[gid]);  // deg >= 1 always
}

// ---------------- fp32 WMMA GEMM: Y[N,128] = X[N,K] @ W[K,128] ----------------
// block = 256 threads (8 waves). Block covers 16 rows; wave w covers cols 16w..16w+15.
__global__ __launch_bounds__(256) void gemm_wmma_kernel(
    const float* __restrict__ X, const float* __restrict__ W,
    float* __restrict__ Y, int K) {
    const int wave = threadIdx.x >> 5;
    const int lane = threadIdx.x & 31;
    const int row0 = blockIdx.x << 4;
    const int col0 = wave << 4;
    const int m    = lane & 15;        // row within tile (A) / col within tile (B)
    const int kh   = (lane >> 4) << 1; // 0 for lanes 0-15, 2 for lanes 16-31

    // A frag (16x4 f32): lanes 0-15 hold K={0,1}, lanes 16-31 hold K={2,3}
    const float* xrow = X + (size_t)(row0 + m) * K + kh;
    // B frag (4x16 f32): same K split across lane halves, row stride = HDIM
    const float* wcol = W + (size_t)kh * HDIM + col0 + m;

    v8f c = {};
    #pragma unroll 4
    for (int k = 0; k < K; k += 4) {
        v2f a, b;
        a.x = xrow[k];
        a.y = xrow[k + 1];
        b.x = wcol[k * HDIM];
        b.y = wcol[k * HDIM + HDIM];
        // D = A(16x4) * B(4x16) + C, fp32 accumulate
        c = __builtin_amdgcn_wmma_f32_16x16x4_f32(
                false, a, false, b, (short)0, c, false, false);
    }

    // C/D layout: VGPR v -> row M=v (lanes 0-15), M=v+8 (lanes 16-31); col = lane&15
    const int mbase = row0 + ((lane >> 4) << 3);
    float* out = Y + (size_t)mbase * HDIM + col0 + m;
    #pragma unroll
    for (int v = 0; v < 8; ++v)
        out[(size_t)v * HDIM] = c[v];
}

// ---------------- zero scratch ----------------
__global__ void zero_kernel(float* p, long n) {
    long gid = (long)blockIdx.x * blockDim.x + threadIdx.x;
    if (gid < n) p[gid] = 0.0f;
}

// ---------------- edge scatter: acc[col] += tmp[row] * dinv[row]*dinv[col] ----------------
// one thread per (edge, float4 chunk); HDIM/4 = 32 chunks per edge
__global__ __launch_bounds__(256) void scatter_kernel(
    const float4* __restrict__ tmp, const int* __restrict__ rows,
    const int* __restrict__ cols, const float* __restrict__ dinv,
    float* __restrict__ acc, int e) {
    long gid = (long)blockIdx.x * blockDim.x + threadIdx.x;
    int eid = (int)(gid >> 5);
    int ch  = (int)(gid & 31);
    if (eid >= e) return;
    int r = rows[eid];
    int cl = cols[eid];
    float nrm = dinv[r] * dinv[cl];
    float4 msg = tmp[(size_t)r * 32 + ch];
    float* dst = acc + (size_t)cl * HDIM + (ch << 2);
    atomicAdd(dst + 0, msg.x * nrm);
    atomicAdd(dst + 1, msg.y * nrm);
    atomicAdd(dst + 2, msg.z * nrm);
    atomicAdd(dst + 3, msg.w * nrm);
}

// ---------------- self-loop + bias + PReLU ----------------
__global__ void act_kernel(const float* __restrict__ acc, const float* __restrict__ tmp,
                           const float* __restrict__ dinv, const float* __restrict__ bias,
                           const float* __restrict__ prelu, float* __restrict__ dst, int n) {
    long gid = (long)blockIdx.x * blockDim.x + threadIdx.x;
    if (gid >= (long)n * HDIM) return;
    int i = (int)(gid >> 7);
    int f = (int)(gid & (HDIM - 1));
    float d = dinv[i];
    float val = acc[gid] + tmp[gid] * d * d + bias[f];
    dst[gid] = (val >= 0.0f) ? val : prelu[f] * val;
}

// ---------------- l2norm both branches + softmax(alpha) blend ----------------
// one wave per row; each lane owns a float4 (32*4 = 128 features)
__global__ __launch_bounds__(256) void finalize_kernel(
    const float* __restrict__ h1, const float* __restrict__ h2,
    const float* __restrict__ alpha, float* __restrict__ out, int n) {
    int wave = threadIdx.x >> 5;
    int lane = threadIdx.x & 31;
    int r = blockIdx.x * 8 + wave;
    if (r >= n) return;

    float a0 = alpha[0], a1 = alpha[1];
    float mx = fmaxf(a0, a1);
    float e0 = __expf(a0 - mx), e1 = __expf(a1 - mx);
    float inv = 1.0f / (e0 + e1);
    float w0 = e0 * inv, w1 = e1 * inv;

    float4 v1 = ((const float4*)(h1 + (size_t)r * HDIM))[lane];
    float4 v2 = ((const float4*)(h2 + (size_t)r * HDIM))[lane];
    float s1 = v1.x * v1.x + v1.y * v1.y + v1.z * v1.z + v1.w * v1.w;
    float s2 = v2.x * v2.x + v2.y * v2.y + v2.z * v2.z + v2.w * v2.w;
    #pragma unroll
    for (int o = 16; o > 0; o >>= 1) {
        s1 += __shfl_xor(s1, o, 32);
        s2 += __shfl_xor(s2, o, 32);
    }
    float i1 = 1.0f / fmaxf(sqrtf(s1), 1e-12f);
    float i2 = 1.0f / fmaxf(sqrtf(s2), 1e-12f);
    float4 o4;
    o4.x = v1.x * i1 * w0 + v2.x * i2 * w1;
    o4.y = v1.y * i1 * w0 + v2.y * i2 * w1;
    o4.z = v1.z * i1 * w0 + v2.z * i2 * w1;
    o4.w = v1.w * i1 * w0 + v2.w * i2 * w1;
    ((float4*)(out + (size_t)r * HDIM))[lane] = o4;
}

static inline int cdiv(long a, int b) { return (int)((a + b - 1) / b); }

extern "C" void kernel_launch(void* const* d_in, const int* in_sizes, int n_in,
                              void* d_out, int out_size, void* d_ws, size_t ws_size,
                              hipStream_t stream) {
    const float* x1    = (const float*)d_in[0];   // [N,256]
    const float* x2    = (const float*)d_in[1];   // [N,128]
    const int*   eidx  = (const int*)d_in[2];     // [2,E]
    const float* W1    = (const float*)d_in[3];
    const float* b1    = (const float*)d_in[4];
    const float* W2    = (const float*)d_in[5];
    const float* b2    = (const float*)d_in[6];
    const float* W3    = (const float*)d_in[7];
    const float* b3    = (const float*)d_in[8];
    const float* W4    = (const float*)d_in[9];
    const float* b4    = (const float*)d_in[10];
    const float* p1    = (const float*)d_in[11];
    const float* p2    = (const float*)d_in[12];
    const float* p3    = (const float*)d_in[13];
    const float* p4    = (const float*)d_in[14];
    const float* alpha = (const float*)d_in[15];

    const int IN1 = 256, IN2 = 128;
    const int N = in_sizes[0] / IN1;  // 50000
    const int E = in_sizes[2] / 2;    // 800000
    const int* row = eidx;
    const int* col = eidx + E;

    // workspace carve-out: dinv | tmp | acc | hA | hB
    float* ws   = (float*)d_ws;
    float* dinv = ws;
    float* tmp  = dinv + N;
    float* acc  = tmp + (size_t)N * HDIM;
    float* hA   = acc + (size_t)N * HDIM;
    float* hB   = hA  + (size_t)N * HDIM;

    const long NH = (long)N * HDIM;

    // D^-1/2 with self-loops
    deg_init_kernel<<<cdiv(N, 256), 256, 0, stream>>>(dinv, N);
    deg_accum_kernel<<<cdiv(E, 256), 256, 0, stream>>>(col, dinv, E);
    deg_rsqrt_kernel<<<cdiv(N, 256), 256, 0, stream>>>(dinv, N);

    auto layer = [&](const float* X, int K, const float* W, const float* b,
                     const float* p, float* dst) {
        gemm_wmma_kernel<<<N / 16, 256, 0, stream>>>(X, W, tmp, K);
        zero_kernel<<<cdiv(NH, 256), 256, 0, stream>>>(acc, NH);
        scatter_kernel<<<cdiv((long)E * 32, 256), 256, 0, stream>>>(
            (const float4*)tmp, row, col, dinv, acc, E);
        act_kernel<<<cdiv(NH, 256), 256, 0, stream>>>(acc, tmp, dinv, b, p, dst, N);
    };

    // branch 1: x1 -> W1 -> W3
    layer(x1, IN1, W1, b1, p1, hA);
    layer(hA, HDIM, W3, b3, p2, hA);
    // branch 2: x2 -> W2 -> W4
    layer(x2, IN2, W2, b2, p3, hB);
    layer(hB, HDIM, W4, b4, p4, hB);

    finalize_kernel<<<cdiv(N, 8), 256, 0, stream>>>(hA, hB, alpha, (float*)d_out, N);
}